// SSITrimmedMAELoss_32229434589420
// MI455X (gfx1250) — compile-verified
//
#include <hip/hip_runtime.h>
#include <stdint.h>

// ---------------- problem constants ----------------
#define BATCH   64
#define HH      512
#define WW      512
#define NPX     (HH * WW)          // 262144 elements per sample
#define SPLIT   16                 // WGs per sample for streaming passes
#define NB      2048               // histogram bins per refinement level
#define TH      8                  // rows per tile (== max subsample step!)
#define NBLK0   (HH / TH)          // 64 row-blocks per sample
#define TRIM_IDX 209715u           // int((1-0.2) * 262144)
#define ALPHA   0.5f

typedef __attribute__((ext_vector_type(2))) float v2f;
typedef __attribute__((ext_vector_type(8))) float v8f;

#if __has_builtin(__builtin_amdgcn_wmma_f32_16x16x4_f32)
#define USE_WMMA 1
#else
#define USE_WMMA 0
#endif

#if __has_builtin(__builtin_amdgcn_tensor_load_to_lds) && __has_builtin(__builtin_amdgcn_s_wait_tensorcnt)
#define USE_TDM 1
#else
#define USE_TDM 0
#endif

// ---------------- workspace state ----------------
struct State {
  unsigned gh_p[BATCH][NB];        // global histogram (pred / residual)
  unsigned gh_t[BATCH][NB];        // global histogram (tgt)
  unsigned n_part[BATCH][SPLIT];
  unsigned n[BATCH];
  unsigned selp_prefix[BATCH], selp_rank[BATCH];
  unsigned selt_prefix[BATCH], selt_rank[BATCH];
  unsigned selr_prefix[BATCH], selr_rank[BATCH];
  float shift_p[BATCH], shift_t[BATCH];
  float scale_p[BATCH], scale_t[BATCH];
  float thr[BATCH];
  float sc_part_p[BATCH][SPLIT], sc_part_t[BATCH][SPLIT];
  // fused tile-kernel partials (deterministic, no global float atomics)
  float mae0part[BATCH][NBLK0];
  float g0part[BATCH][NBLK0];
  float gspart[3][BATCH][NBLK0];   // gradient sums, scales 1..3
  float nspart[3][BATCH][NBLK0];   // mask counts,   scales 1..3
};

// ---------------- helpers ----------------
__device__ __forceinline__ unsigned f2ord(float f) {
  unsigned u = __float_as_uint(f);
  return (u & 0x80000000u) ? ~u : (u | 0x80000000u);
}
__device__ __forceinline__ float ord2f(unsigned k) {
  unsigned u = (k & 0x80000000u) ? (k & 0x7FFFFFFFu) : ~k;
  return __uint_as_float(u);
}
// level 0: bits[31:21] (2048 bins), level 1: bits[20:10] (2048), level 2: bits[9:0] (1024)
__device__ __forceinline__ unsigned lvl_mask(int l) {
  return l == 0 ? 0u : (l == 1 ? 0xFFE00000u : 0xFFFFFC00u);
}
__device__ __forceinline__ unsigned lvl_bin(unsigned k, int l) {
  return l == 0 ? (k >> 21) : (l == 1 ? ((k >> 10) & 0x7FFu) : (k & 0x3FFu));
}
__device__ __forceinline__ int lvl_nbins(int l) { return l == 2 ? 1024 : 2048; }

__device__ __forceinline__ void select_hist(const unsigned* h, int nbins, unsigned rank,
                                            unsigned* bin, unsigned* rem) {
  unsigned cum = 0u;
  for (int i = 0; i < nbins; ++i) {
    unsigned c = h[i];
    if (rank < cum + c) { *bin = (unsigned)i; *rem = rank - cum; return; }
    cum += c;
  }
  *bin = (unsigned)(nbins - 1); *rem = 0u;
}

// Wave32 sum reduction through the matrix pipe (XDL co-executes with VALU):
// A (16x4 f32) holds the 32 per-lane partials in K=0 (lanes 0-15) / K=2 (lanes 16-31),
// B = ones -> D row-sums pair lanes L and L+16; summing the 8 D VGPRs per lane and one
// shfl_xor(16) yields the full wave sum in every lane.
__device__ __forceinline__ float waveReduceSum(float v) {
#if USE_WMMA
  v2f a = {v, 0.0f};
  v2f bo = {1.0f, 1.0f};
  v8f c = {0.f, 0.f, 0.f, 0.f, 0.f, 0.f, 0.f, 0.f};
  c = __builtin_amdgcn_wmma_f32_16x16x4_f32(false, a, false, bo, (short)0, c, false, false);
  float s = c[0] + c[1] + c[2] + c[3] + c[4] + c[5] + c[6] + c[7];
  s += __shfl_xor(s, 16, 32);
  return s;
#else
  for (int off = 16; off > 0; off >>= 1) v += __shfl_xor(v, off, 32);
  return v;
#endif
}

__device__ __forceinline__ float blockReduceSum(float v, float* red) {
  float w = waveReduceSum(v);
  int lane = (int)threadIdx.x & 31;
  int wid  = (int)threadIdx.x >> 5;
  if (lane == 0) red[wid] = w;
  __syncthreads();
  float r = 0.f;
  if (threadIdx.x == 0) {
    int nw = ((int)blockDim.x + 31) >> 5;
    for (int i = 0; i < nw; ++i) r += red[i];
  }
  __syncthreads();
  return r;  // valid on thread 0
}

// ---------------- TDM tile load (2D, f32) ----------------
#if USE_TDM
typedef unsigned int u32x4_t __attribute__((ext_vector_type(4)));
typedef int i32x4_t __attribute__((ext_vector_type(4)));
typedef int i32x8_t __attribute__((ext_vector_type(8)));

__device__ __forceinline__ unsigned lds_off(const void* p) {
  // LDS aperture: low 32 bits of the generic address are the LDS byte offset.
  return (unsigned)(uintptr_t)p;
}

__device__ __forceinline__ void tdm_load_tile_f32(const void* gptr, unsigned ldsb,
                                                  unsigned tensor_w, unsigned tensor_h,
                                                  unsigned tile_w, unsigned tile_h,
                                                  unsigned row_stride_elems) {
  unsigned long long ga = (unsigned long long)(uintptr_t)gptr;
  u32x4_t g0;
  g0[0] = 1u;                                             // count=1, user mode
  g0[1] = ldsb;                                           // lds_addr (bytes)
  g0[2] = (unsigned)(ga & 0xFFFFFFFFu);                   // global_addr[31:0]
  g0[3] = (unsigned)((ga >> 32) & 0x01FFFFFFu) | (2u << 30); // addr[56:32] | type=2
  i32x8_t g1;
  g1[0] = (int)(2u << 16);                                // data_size=4B, no multicast
  g1[1] = (int)((tensor_w & 0xFFFFu) << 16);              // tensor_dim0[15:0]
  g1[2] = (int)(((tensor_w >> 16) & 0xFFFFu) | ((tensor_h & 0xFFFFu) << 16));
  g1[3] = (int)(((tensor_h >> 16) & 0xFFFFu) | ((tile_w & 0xFFFFu) << 16));
  g1[4] = (int)(tile_h & 0xFFFFu);                        // tile_dim1, tile_dim2=0
  g1[5] = (int)row_stride_elems;                          // tensor_dim0_stride[31:0]
  g1[6] = 0;
  g1[7] = 0;
  i32x4_t z4 = {0, 0, 0, 0};
#if defined(__clang_major__) && (__clang_major__ >= 23)
  i32x8_t z8 = {0, 0, 0, 0, 0, 0, 0, 0};
  __builtin_amdgcn_tensor_load_to_lds(g0, g1, z4, z4, z8, 0);
#else
  __builtin_amdgcn_tensor_load_to_lds(g0, g1, z4, z4, 0);
#endif
}
#endif  // USE_TDM

// ---------------- kernels ----------------

// zero the global histograms once per launch (ws is poisoned by the harness)
__global__ void k_zero(State* st) {
  unsigned* g = &st->gh_p[0][0];
  size_t total = (size_t)BATCH * NB * 2;
  for (size_t i = (size_t)blockIdx.x * blockDim.x + threadIdx.x; i < total;
       i += (size_t)gridDim.x * blockDim.x)
    g[i] = 0u;
}

// one radix-select refinement level for both medians (pred & tgt), masked by tgt>0
__global__ void k_hist_median(const float* __restrict__ pred, const float* __restrict__ tgt,
                              State* st, int level) {
  __shared__ unsigned hp[NB];
  __shared__ unsigned ht[NB];
  __shared__ unsigned ncnt;
  const int b = blockIdx.x, chunk = blockIdx.y;
  for (int i = threadIdx.x; i < NB; i += blockDim.x) { hp[i] = 0u; ht[i] = 0u; }
  if (threadIdx.x == 0) ncnt = 0u;
  __syncthreads();

  const unsigned msk = lvl_mask(level);
  const unsigned prefP = st->selp_prefix[b] & msk;
  const unsigned prefT = st->selt_prefix[b] & msk;
  const float4* p4 = reinterpret_cast<const float4*>(pred) + (size_t)b * (NPX / 4);
  const float4* t4 = reinterpret_cast<const float4*>(tgt)  + (size_t)b * (NPX / 4);
  const int per = (NPX / 4) / SPLIT;
  const int base = chunk * per;

  unsigned myn = 0u;
  for (int i = threadIdx.x; i < per; i += blockDim.x) {
    const int idx = base + i;
    __builtin_prefetch((const void*)(p4 + idx + 1024), 0, 0);
    __builtin_prefetch((const void*)(t4 + idx + 1024), 0, 0);
    float4 pv = p4[idx];
    float4 tv = t4[idx];
    float pe[4] = {pv.x, pv.y, pv.z, pv.w};
    float te[4] = {tv.x, tv.y, tv.z, tv.w};
#pragma unroll
    for (int j = 0; j < 4; ++j) {
      if (te[j] > 0.0f) {
        ++myn;
        unsigned kp = f2ord(pe[j]);
        unsigned kt = f2ord(te[j]);
        if ((kp & msk) == prefP) atomicAdd(&hp[lvl_bin(kp, level)], 1u);
        if ((kt & msk) == prefT) atomicAdd(&ht[lvl_bin(kt, level)], 1u);
      }
    }
  }
  if (level == 0 && myn) atomicAdd(&ncnt, myn);
  __syncthreads();
  if (level == 0 && threadIdx.x == 0) st->n_part[b][chunk] = ncnt;
  for (int i = threadIdx.x; i < NB; i += blockDim.x) {
    if (hp[i]) atomicAdd(&st->gh_p[b][i], hp[i]);
    if (ht[i]) atomicAdd(&st->gh_t[b][i], ht[i]);
  }
}

__global__ void k_select_median(State* st, int level) {
  const int b = blockIdx.x;
  if (threadIdx.x == 0) {
    unsigned n;
    if (level == 0) {
      n = 0u;
      for (int i = 0; i < SPLIT; ++i) n += st->n_part[b][i];
      st->n[b] = n;
    } else {
      n = st->n[b];
    }
    const int nb = lvl_nbins(level);
    unsigned bin, rem, rank;
    // pred median
    rank = (level == 0) ? (n ? (n - 1u) / 2u : 0u) : st->selp_rank[b];
    select_hist(&st->gh_p[b][0], nb, rank, &bin, &rem);
    if (level == 0)      { st->selp_prefix[b] = bin << 21;  st->selp_rank[b] = rem; }
    else if (level == 1) { st->selp_prefix[b] |= bin << 10; st->selp_rank[b] = rem; }
    else                 { st->shift_p[b] = n ? ord2f(st->selp_prefix[b] | bin) : 0.0f; }
    // tgt median
    rank = (level == 0) ? (n ? (n - 1u) / 2u : 0u) : st->selt_rank[b];
    select_hist(&st->gh_t[b][0], nb, rank, &bin, &rem);
    if (level == 0)      { st->selt_prefix[b] = bin << 21;  st->selt_rank[b] = rem; }
    else if (level == 1) { st->selt_prefix[b] |= bin << 10; st->selt_rank[b] = rem; }
    else                 { st->shift_t[b] = n ? ord2f(st->selt_prefix[b] | bin) : 0.0f; }
  }
  __syncthreads();
  for (int i = threadIdx.x; i < NB; i += blockDim.x) {
    st->gh_p[b][i] = 0u;
    st->gh_t[b][i] = 0u;
  }
}

// scale = mean |x - shift| over valid pixels
__global__ void k_scale_part(const float* __restrict__ pred, const float* __restrict__ tgt,
                             State* st) {
  const int b = blockIdx.x, chunk = blockIdx.y;
  const float shp = st->shift_p[b], sht = st->shift_t[b];
  const float4* p4 = reinterpret_cast<const float4*>(pred) + (size_t)b * (NPX / 4);
  const float4* t4 = reinterpret_cast<const float4*>(tgt)  + (size_t)b * (NPX / 4);
  const int per = (NPX / 4) / SPLIT;
  const int base = chunk * per;
  float ap = 0.f, at = 0.f;
  for (int i = threadIdx.x; i < per; i += blockDim.x) {
    const int idx = base + i;
    float4 pv = p4[idx];
    float4 tv = t4[idx];
    float pe[4] = {pv.x, pv.y, pv.z, pv.w};
    float te[4] = {tv.x, tv.y, tv.z, tv.w};
#pragma unroll
    for (int j = 0; j < 4; ++j) {
      if (te[j] > 0.0f) {
        ap += fabsf(pe[j] - shp);
        at += fabsf(te[j] - sht);
      }
    }
  }
  __shared__ float r1[8];
  __shared__ float r2[8];
  float bp = blockReduceSum(ap, r1);
  float bt = blockReduceSum(at, r2);
  if (threadIdx.x == 0) { st->sc_part_p[b][chunk] = bp; st->sc_part_t[b][chunk] = bt; }
}

__global__ void k_scale_fin(State* st) {
  const int b = blockIdx.x;
  if (threadIdx.x == 0) {
    float sp = 0.f, stt = 0.f;
    for (int i = 0; i < SPLIT; ++i) { sp += st->sc_part_p[b][i]; stt += st->sc_part_t[b][i]; }
    unsigned n = st->n[b];
    float dn = (float)(n > 0u ? n : 1u);
    float scp = sp / dn;  if (scp == 0.f) scp = 1.f;
    float sct = stt / dn; if (sct == 0.f) sct = 1.f;
    st->scale_p[b] = scp;
    st->scale_t[b] = sct;
  }
}

// radix-select refinement for the trim threshold on residuals (non-neg floats)
__global__ void k_hist_thr(const float* __restrict__ pred, const float* __restrict__ tgt,
                           State* st, int level) {
  __shared__ unsigned hr[NB];
  const int b = blockIdx.x, chunk = blockIdx.y;
  for (int i = threadIdx.x; i < NB; i += blockDim.x) hr[i] = 0u;
  __syncthreads();

  const unsigned msk = lvl_mask(level);
  const unsigned pref = st->selr_prefix[b] & msk;
  const float shp = st->shift_p[b], sht = st->shift_t[b];
  const float icp = 1.f / st->scale_p[b], ict = 1.f / st->scale_t[b];
  const float4* p4 = reinterpret_cast<const float4*>(pred) + (size_t)b * (NPX / 4);
  const float4* t4 = reinterpret_cast<const float4*>(tgt)  + (size_t)b * (NPX / 4);
  const int per = (NPX / 4) / SPLIT;
  const int base = chunk * per;

  for (int i = threadIdx.x; i < per; i += blockDim.x) {
    const int idx = base + i;
    __builtin_prefetch((const void*)(p4 + idx + 1024), 0, 0);
    __builtin_prefetch((const void*)(t4 + idx + 1024), 0, 0);
    float4 pv = p4[idx];
    float4 tv = t4[idx];
    float pe[4] = {pv.x, pv.y, pv.z, pv.w};
    float te[4] = {tv.x, tv.y, tv.z, tv.w};
#pragma unroll
    for (int j = 0; j < 4; ++j) {
      unsigned key = (te[j] > 0.0f)
          ? __float_as_uint(fabsf((pe[j] - shp) * icp - (te[j] - sht) * ict))
          : 0u;
      if ((key & msk) == pref) atomicAdd(&hr[lvl_bin(key, level)], 1u);
    }
  }
  __syncthreads();
  for (int i = threadIdx.x; i < NB; i += blockDim.x)
    if (hr[i]) atomicAdd(&st->gh_p[b][i], hr[i]);
}

__global__ void k_select_thr(State* st, int level) {
  const int b = blockIdx.x;
  if (threadIdx.x == 0) {
    unsigned rank;
    if (level == 0) {
      unsigned long long r = (unsigned long long)(NPX - st->n[b]) + TRIM_IDX;
      rank = (r > (NPX - 1)) ? (unsigned)(NPX - 1) : (unsigned)r;
    } else {
      rank = st->selr_rank[b];
    }
    unsigned bin, rem;
    select_hist(&st->gh_p[b][0], lvl_nbins(level), rank, &bin, &rem);
    if (level == 0)      { st->selr_prefix[b] = bin << 21;  st->selr_rank[b] = rem; }
    else if (level == 1) { st->selr_prefix[b] |= bin << 10; st->selr_rank[b] = rem; }
    else                 { st->thr[b] = __uint_as_float(st->selr_prefix[b] | bin); }
  }
  __syncthreads();
  for (int i = threadIdx.x; i < NB; i += blockDim.x) st->gh_p[b][i] = 0u;
}

// ---------------------------------------------------------------------------
// Fused tail kernel: TDM stages a (TH+1)x512 tile of pred & tgt into LDS, then
// computes (a) the trimmed-MAE partial sum and (b) gradient-loss partials for
// ALL four scales from the same tile (TH == max step, so every scale-s
// neighbor lies inside the tile). d = pred_n - tgt_n is computed once,
// in place over the pred tile. Each pixel is read once; halo rows by two blocks.
// ---------------------------------------------------------------------------
__global__ void k_tile_loss(const float* __restrict__ pred, const float* __restrict__ tgt,
                            State* st) {
  __shared__ __align__(128) float sp[(TH + 1) * WW];   // pred tile -> d tile
  __shared__ __align__(128) float stg[(TH + 1) * WW];  // tgt tile (mask source)
  __shared__ float red[8];
  const int b = blockIdx.x;
  const int r0 = blockIdx.y * TH;
  const int rows_avail = HH - r0;
  const int nrows = rows_avail < (TH + 1) ? rows_avail : (TH + 1);
  const float* pbase = pred + (size_t)b * NPX + (size_t)r0 * WW;
  const float* tbase = tgt  + (size_t)b * NPX + (size_t)r0 * WW;

#if USE_TDM
  if (threadIdx.x < 32) {
    // tensor_dim1 = remaining rows: TDM OOB rows fill zeros -> masked out anyway
    tdm_load_tile_f32(pbase, lds_off(sp),  WW, (unsigned)rows_avail, WW, TH + 1, WW);
    tdm_load_tile_f32(tbase, lds_off(stg), WW, (unsigned)rows_avail, WW, TH + 1, WW);
    __builtin_amdgcn_s_wait_tensorcnt(0);
  }
  __syncthreads();
#else
  for (int i = threadIdx.x; i < nrows * WW; i += blockDim.x) {
    sp[i] = pbase[i];
    stg[i] = tbase[i];
  }
  __syncthreads();
#endif

  const float shp = st->shift_p[b], sht = st->shift_t[b];
  const float icp = 1.f / st->scale_p[b], ict = 1.f / st->scale_t[b];
  const float thr = st->thr[b];

  // normalize once: sp <- d = pred_n - tgt_n
  for (int i = threadIdx.x; i < nrows * WW; i += blockDim.x) {
    sp[i] = (sp[i] - shp) * icp - (stg[i] - sht) * ict;
  }
  __syncthreads();

  // ---- dense pass: trimmed MAE + scale-0 gradient ----
  float mae_acc = 0.f;
  float g0_acc = 0.f;
  for (int idx = threadIdx.x; idx < TH * WW; idx += blockDim.x) {
    const int lr = idx >> 9;          // WW == 512
    const int c = idx & (WW - 1);
    const bool m0 = stg[idx] > 0.f;
    const float d0 = sp[idx];
    if (m0) {
      float a = fabsf(d0);
      if (a <= thr) mae_acc += a;
    }
    if (c + 1 < WW && m0 && stg[idx + 1] > 0.f)
      g0_acc += fabsf(sp[idx + 1] - d0);
    if (r0 + lr + 1 < HH && m0 && stg[idx + WW] > 0.f)
      g0_acc += fabsf(sp[idx + WW] - d0);
  }
  float bs = blockReduceSum(mae_acc, red);
  if (threadIdx.x == 0) st->mae0part[b][blockIdx.y] = bs;
  bs = blockReduceSum(g0_acc, red);
  if (threadIdx.x == 0) st->g0part[b][blockIdx.y] = bs;

  // ---- sub-sampled scales 1..3 from the same tile ----
#pragma unroll
  for (int s = 1; s < 4; ++s) {
    const int stp = 1 << s;
    const int nr = TH >> s;           // grid rows owned by this block
    const int nc = WW >> s;           // grid cols
    const int tot = nr * nc;
    float acc = 0.f, cnt = 0.f;
    for (int k = threadIdx.x; k < tot; k += blockDim.x) {
      const int ir = k / nc, j = k - ir * nc;
      const int lr = ir * stp, c = j * stp;
      const int idx = lr * WW + c;
      const bool m0 = stg[idx] > 0.f;
      if (m0) cnt += 1.f;
      const float d0 = sp[idx];
      if (c + stp < WW && m0 && stg[idx + stp] > 0.f)
        acc += fabsf(sp[idx + stp] - d0);
      if (r0 + lr + stp < HH && m0 && stg[idx + stp * WW] > 0.f)
        acc += fabsf(sp[idx + stp * WW] - d0);
    }
    float gs = blockReduceSum(acc, red);
    float ns = blockReduceSum(cnt, red);
    if (threadIdx.x == 0) {
      st->gspart[s - 1][b][blockIdx.y] = gs;
      st->nspart[s - 1][b][blockIdx.y] = ns;
    }
  }
}

__global__ void k_finalize(State* st, float* out) {
  __shared__ float acc[BATCH];
  const int b = (int)threadIdx.x;
  if (b < BATCH) {
    unsigned n = st->n[b];
    float msum = 0.f, g0 = 0.f;
    for (int i = 0; i < NBLK0; ++i) { msum += st->mae0part[b][i]; g0 += st->g0part[b][i]; }
    float mae = (n > 0u) ? msum / (float)(2u * n) : 0.f;
    float gl = (n > 0u) ? g0 / (float)n : 0.f;   // scale-0 norm == n (mask.sum)
    for (int s = 0; s < 3; ++s) {
      float gs = 0.f, ns = 0.f;
      for (int i = 0; i < NBLK0; ++i) { gs += st->gspart[s][b][i]; ns += st->nspart[s][b][i]; }
      gl += (ns > 0.f) ? gs / ns : 0.f;
    }
    acc[b] = mae + ALPHA * gl;
  }
  __syncthreads();
  if (b == 0) {
    float s2 = 0.f;
    for (int i = 0; i < BATCH; ++i) s2 += acc[i];
    out[0] = s2 / (float)BATCH;
  }
}

// ---------------- launcher ----------------
extern "C" void kernel_launch(void* const* d_in, const int* in_sizes, int n_in,
                              void* d_out, int out_size, void* d_ws, size_t ws_size,
                              hipStream_t stream) {
  (void)in_sizes; (void)n_in; (void)out_size; (void)ws_size;
  const float* pred = (const float*)d_in[0];
  const float* tgt  = (const float*)d_in[1];
  State* st = (State*)d_ws;
  float* out = (float*)d_out;

  dim3 blk(256);
  k_zero<<<256, 256, 0, stream>>>(st);
  for (int lvl = 0; lvl < 3; ++lvl) {
    k_hist_median<<<dim3(BATCH, SPLIT), blk, 0, stream>>>(pred, tgt, st, lvl);
    k_select_median<<<BATCH, blk, 0, stream>>>(st, lvl);
  }
  k_scale_part<<<dim3(BATCH, SPLIT), blk, 0, stream>>>(pred, tgt, st);
  k_scale_fin<<<BATCH, 32, 0, stream>>>(st);
  for (int lvl = 0; lvl < 3; ++lvl) {
    k_hist_thr<<<dim3(BATCH, SPLIT), blk, 0, stream>>>(pred, tgt, st, lvl);
    k_select_thr<<<BATCH, blk, 0, stream>>>(st, lvl);
  }
  k_tile_loss<<<dim3(BATCH, NBLK0), blk, 0, stream>>>(pred, tgt, st);
  k_finalize<<<1, 64, 0, stream>>>(st, out);
}